// XposHead_32358283608665
// MI455X (gfx1250) — compile-verified
//
#include <hip/hip_runtime.h>
#include <hip/hip_bf16.h>

typedef __attribute__((ext_vector_type(2))) float v2f;
typedef __attribute__((ext_vector_type(8))) float v8f;

#define L_   13
#define B_   32
#define S_   512
#define D_   768
#define T_   256
#define NT_  50
#define TOK  254               // tokens that get logits (mixed[:, :-1])
#define M_TOT (B_ * TOK)       // 8128, divisible by 16
#define NPAD 64                // padded N for WMMA tiles
#define KCHUNK 192             // K-chunk staged in LDS: 192*64*4 = 48KB
#define D4   (D_ / 4)          // 192 float4 per row

// ---------------------------------------------------------------- kernel 0
// token_counts[b] = (#rows of token_spans[b] with both entries > 0) - 1
__global__ void k_counts(const int* __restrict__ spans, int* __restrict__ tc) {
    int b = blockIdx.x;
    int t = threadIdx.x;
    __shared__ int cnt;
    if (t == 0) cnt = 0;
    __syncthreads();
    if (t < T_) {
        int s0 = spans[(b * T_ + t) * 2 + 0];
        int s1 = spans[(b * T_ + t) * 2 + 1];
        if (s0 > 0 && s1 > 0) atomicAdd(&cnt, 1);
    }
    __syncthreads();
    if (t == 0) tc[b] = cnt - 1;
}

// ---------------------------------------------------------------- kernel 1
// Fused span-pool + layer softmax-mix (the roofline kernel: streams 654 MB):
// tok[b,t,:] = gamma * sum_l wmix[l] * mean_{s in span(b,t+1)} hidden[l,b,s,:]
// One block per (t, b); 192 threads, one float4 (global_load_b128) per row
// per thread -> each needed hidden row is read exactly once, fully coalesced.
__global__ void k_pool_mix(const float* __restrict__ hidden,
                           const int*   __restrict__ spans,
                           const int*   __restrict__ tc,
                           const float* __restrict__ scalars,
                           const float* __restrict__ gamma,
                           float*       __restrict__ tok) {
    int t   = blockIdx.x;            // 0..TOK-1
    int b   = blockIdx.y;            // 0..B-1
    int tid = threadIdx.x;           // 0..191 (D4)

    __shared__ float wmix[L_];
    __shared__ float g;
    if (tid == 0) {
        float mx = scalars[0];
        for (int l = 1; l < L_; ++l) mx = fmaxf(mx, scalars[l]);
        float s = 0.f;
        for (int l = 0; l < L_; ++l) { float e = expf(scalars[l] - mx); wmix[l] = e; s += e; }
        float inv = 1.f / s;
        for (int l = 0; l < L_; ++l) wmix[l] *= inv;
        g = gamma[0];
    }
    __syncthreads();

    int start = spans[(b * T_ + t + 1) * 2 + 0];
    int end   = spans[(b * T_ + t + 1) * 2 + 1];
    if (t == tc[b]) { start = 0; end = 0; }       // zero_row logic
    int cntv    = end - start;
    float scale = g / (float)max(cntv, 1);

    float4 acc = make_float4(0.f, 0.f, 0.f, 0.f);
    for (int l = 0; l < L_; ++l) {
        float wl = wmix[l];
        size_t basel = ((size_t)l * B_ + b) * S_;
        for (int s = start; s < end; ++s) {
            const float4* row = (const float4*)(hidden + (basel + (size_t)s) * D_);
            float4 h = row[tid];                 // global_load_b128, coalesced
            acc.x = fmaf(wl, h.x, acc.x);
            acc.y = fmaf(wl, h.y, acc.y);
            acc.z = fmaf(wl, h.z, acc.z);
            acc.w = fmaf(wl, h.w, acc.w);
        }
    }
    float4* o = (float4*)(tok + ((size_t)b * TOK + t) * D_);
    o[tid] = make_float4(scale * acc.x, scale * acc.y, scale * acc.z, scale * acc.w);
}

// ---------------------------------------------------------------- kernel 2
// logits = tok(8128x768) @ W^T(768x50) + b  via V_WMMA_F32_16X16X4_F32.
// One wave per 16-row M tile; N padded to 64 (4 accumulators).
// W staged transposed into LDS as Wt[k][64] (zero-padded n>=50), 48KB chunks,
// so B-fragment loads (lane n = lane&15 contiguous) are bank-conflict-free.
__global__ void k_gemm_wmma(const float* __restrict__ tok,
                            const float* __restrict__ W,
                            const float* __restrict__ bias,
                            float*       __restrict__ logits) {
    __shared__ float Wt[KCHUNK * NPAD];   // 48 KB
    int lane  = threadIdx.x;              // 0..31 (wave32)
    int mtile = blockIdx.x;               // 0..507
    int half  = lane >> 4;                // 0/1
    int l16   = lane & 15;

    int m = mtile * 16 + l16;
    const float* arow = tok + (size_t)m * D_;

    v8f acc0 = {}, acc1 = {}, acc2 = {}, acc3 = {};

    for (int kc = 0; kc < D_; kc += KCHUNK) {
        // stage W chunk transposed + padded into LDS (single wave -> in-order DS)
        for (int i = lane; i < KCHUNK * NPAD; i += 32) Wt[i] = 0.f;
        __syncthreads();
        for (int i = lane; i < NT_ * KCHUNK; i += 32) {
            int n = i / KCHUNK, kl = i % KCHUNK;
            Wt[kl * NPAD + n] = W[n * D_ + kc + kl];
        }
        __syncthreads();

        for (int k0 = 0; k0 < KCHUNK; k0 += 4) {
            int kk = k0 + half * 2;               // A/B lane K base (ISA 7.12.2)
            // A fragment: a[j] = A[m][kc + kk + j]
            v2f a; a.x = arow[kc + kk]; a.y = arow[kc + kk + 1];
            const float* w0 = &Wt[kk * NPAD];
            const float* w1 = &Wt[(kk + 1) * NPAD];
            v2f b0; b0.x = w0[l16];      b0.y = w1[l16];
            v2f b1; b1.x = w0[16 + l16]; b1.y = w1[16 + l16];
            v2f b2; b2.x = w0[32 + l16]; b2.y = w1[32 + l16];
            v2f b3; b3.x = w0[48 + l16]; b3.y = w1[48 + l16];
            acc0 = __builtin_amdgcn_wmma_f32_16x16x4_f32(false, a, false, b0, (short)0, acc0, false, false);
            acc1 = __builtin_amdgcn_wmma_f32_16x16x4_f32(false, a, false, b1, (short)0, acc1, false, false);
            acc2 = __builtin_amdgcn_wmma_f32_16x16x4_f32(false, a, false, b2, (short)0, acc2, false, false);
            acc3 = __builtin_amdgcn_wmma_f32_16x16x4_f32(false, a, false, b3, (short)0, acc3, false, false);
        }
        __syncthreads();
    }

    // Epilogue: C/D layout -> row M = j + 8*half, col N = l16 (+tile offset)
    for (int j = 0; j < 8; ++j) {
        int mrow = mtile * 16 + j + 8 * half;
        float* orow = logits + (size_t)mrow * NT_;
        int n0 = l16;                              orow[n0] = acc0[j] + bias[n0];
        int n1 = 16 + l16;                         orow[n1] = acc1[j] + bias[n1];
        int n2 = 32 + l16; if (n2 < NT_)           orow[n2] = acc2[j] + bias[n2];
        int n3 = 48 + l16; if (n3 < NT_)           orow[n3] = acc3[j] + bias[n3];
    }
}

// ---------------------------------------------------------------- kernel 3
// Per-batch: preds (argmax * mask, stored as float) + masked mean NLL.
__global__ void k_loss_batch(const float* __restrict__ logits,
                             const int*   __restrict__ tags,
                             const int*   __restrict__ tc,
                             float*       __restrict__ preds,
                             float*       __restrict__ per_batch,
                             float*       __restrict__ nonempty) {
    int b = blockIdx.x;
    int t = threadIdx.x;                 // 256 threads, tokens 0..253
    __shared__ float s_nll[256], s_w[256];
    float nll = 0.f, w = 0.f;
    if (t < TOK) {
        const float* row = logits + ((size_t)b * TOK + t) * NT_;
        float m = row[0]; int am = 0;
        for (int n = 1; n < NT_; ++n) { float v = row[n]; if (v > m) { m = v; am = n; } }
        float se = 0.f;
        for (int n = 0; n < NT_; ++n) se += expf(row[n] - m);
        float logZ = m + logf(se);
        int tag = tags[b * TOK + t];
        int mask = (t < tc[b]) ? 1 : 0;
        preds[b * TOK + t] = mask ? (float)am : 0.f;
        nll = (logZ - row[tag]) * (float)mask;
        w   = (float)mask;
    }
    s_nll[t] = nll; s_w[t] = w;
    __syncthreads();
    for (int off = 128; off > 0; off >>= 1) {
        if (t < off) { s_nll[t] += s_nll[t + off]; s_w[t] += s_w[t + off]; }
        __syncthreads();
    }
    if (t == 0) {
        per_batch[b] = s_nll[0] / (s_w[0] + 1e-13f);
        nonempty[b]  = (s_w[0] > 0.f) ? 1.f : 0.f;
    }
}

__global__ void k_loss_final(const float* __restrict__ per_batch,
                             const float* __restrict__ nonempty,
                             float*       __restrict__ loss) {
    if (threadIdx.x == 0) {
        float s = 0.f, ne = 0.f;
        for (int b = 0; b < B_; ++b) { s += per_batch[b]; ne += nonempty[b]; }
        loss[0] = s / (ne + 1e-13f);
    }
}

// ---------------------------------------------------------------- launch
extern "C" void kernel_launch(void* const* d_in, const int* in_sizes, int n_in,
                              void* d_out, int out_size, void* d_ws, size_t ws_size,
                              hipStream_t stream) {
    const float* hidden  = (const float*)d_in[0];
    const int*   spans   = (const int*)  d_in[1];
    // d_in[2] tree_is_gold: unused by the reference computation
    const int*   tags    = (const int*)  d_in[3];
    const float* W       = (const float*)d_in[4];
    const float* bias    = (const float*)d_in[5];
    const float* scalars = (const float*)d_in[6];
    const float* gamma   = (const float*)d_in[7];

    float* logits = (float*)d_out;                       // 8128*50
    float* preds  = logits + (size_t)M_TOT * NT_;        // 8128
    float* loss   = preds + M_TOT;                       // 1

    char* ws = (char*)d_ws;
    float* tok = (float*)ws;                             // 8128*768 floats (~25 MB)
    size_t off = (size_t)M_TOT * D_ * sizeof(float);
    int*   tc        = (int*)(ws + off);   off += B_ * sizeof(int);
    float* per_batch = (float*)(ws + off); off += B_ * sizeof(float);
    float* nonempty  = (float*)(ws + off);

    k_counts<<<B_, 256, 0, stream>>>(spans, tc);
    dim3 g1(TOK, B_);
    k_pool_mix<<<g1, D4, 0, stream>>>(hidden, spans, tc, scalars, gamma, tok);
    k_gemm_wmma<<<M_TOT / 16, 32, 0, stream>>>(tok, W, bias, logits);
    k_loss_batch<<<B_, 256, 0, stream>>>(logits, tags, tc, preds, per_batch, nonempty);
    k_loss_final<<<1, 32, 0, stream>>>(per_batch, nonempty, loss);
}